// ExpAttn_34016140984468
// MI455X (gfx1250) — compile-verified
//
#include <hip/hip_runtime.h>
#include <hip/hip_bf16.h>

#define T_LEN 32
#define E_LEN 256
#define BATCH 16
#define DIM   512
#define HID   512

typedef __attribute__((ext_vector_type(2))) float v2f;
typedef __attribute__((ext_vector_type(8))) float v8f;

// ---------------------------------------------------------------------------
// Batched fp32 WMMA GEMM, templated on B-access mode; one wave computes a
// 16x64 C strip (4 N-tiles): each A fragment feeds 4 V_WMMA_F32_16X16X4_F32.
// Software-pipelined: next k-step's fragments are loaded before the current
// WMMA group consumes its fragments, so loads overlap matrix-core work.
//   C[b][m][n] = sum_k A[b][m][k] * B'[b][k][n]
//   B'[k][n] = BTRANS ? Bm[n*ldb + koff + k] : Bm[k*ldb + n]
// ---------------------------------------------------------------------------
template <int BTRANS>
__global__ void wmma_gemm_f32(const float* __restrict__ A, long long strideAb, int lda,
                              const float* __restrict__ Bm, long long strideBb, int ldb,
                              int koff,
                              float* __restrict__ C, long long strideCb, int ldc,
                              int tilesM, int tilesN4, int K, int batches)
{
    int waveId = (int)((blockIdx.x * blockDim.x + threadIdx.x) >> 5);
    int total  = tilesM * tilesN4 * batches;
    if (waveId >= total) return;              // whole wave exits together

    int b   = waveId / (tilesM * tilesN4);
    int t   = waveId % (tilesM * tilesN4);
    int tm  = t / tilesN4;
    int tn4 = t % tilesN4;                    // strip of 4 consecutive N tiles

    const float* Ab = A  + (long long)b * strideAb + (long long)tm * 16 * lda;
    const float* Bb = Bm + (long long)b * strideBb;
    float*       Cb = C  + (long long)b * strideCb + (long long)tm * 16 * ldc + tn4 * 64;

    int lane  = threadIdx.x & 31;
    int half  = lane >> 4;       // selects K pair {2h, 2h+1}
    int idx   = lane & 15;       // M row for A, N col within tile for B/C
    int n0    = tn4 * 64 + idx;  // global N of tile-0 column
    int half2 = half * 2;

    // Fragment pointers (advanced by 4 K each step)
    const float* ap = Ab + (long long)idx * lda + half2;
    const float* bp[4];
#pragma unroll
    for (int j = 0; j < 4; ++j) {
        if (BTRANS)
            bp[j] = Bb + (long long)(n0 + j * 16) * ldb + koff + half2;
        else
            bp[j] = Bb + (long long)half2 * ldb + n0 + j * 16;
    }

    auto loadB = [&](int j) -> v2f {
        v2f r;
        if (BTRANS) {
            r = *(const v2f*)bp[j];
        } else {
            r.x = bp[j][0];
            r.y = bp[j][(long long)ldb];
        }
        return r;
    };

    v8f acc[4] = {};

    // Prologue: fragments for k = 0
    v2f av = *(const v2f*)ap;
    v2f bv[4];
#pragma unroll
    for (int j = 0; j < 4; ++j) bv[j] = loadB(j);

    for (int k = 0; k < K - 4; k += 4) {
        // Issue next k-step's loads first (hide latency behind WMMAs below)
        ap += 4;
        v2f av_n = *(const v2f*)ap;
        v2f bv_n[4];
#pragma unroll
        for (int j = 0; j < 4; ++j) {
            if (BTRANS) bp[j] += 4;
            else        bp[j] += (long long)4 * ldb;
            bv_n[j] = loadB(j);
        }
        // Consume current fragments
#pragma unroll
        for (int j = 0; j < 4; ++j)
            acc[j] = __builtin_amdgcn_wmma_f32_16x16x4_f32(
                         false, av, false, bv[j], (short)0, acc[j], false, false);
        av = av_n;
#pragma unroll
        for (int j = 0; j < 4; ++j) bv[j] = bv_n[j];
    }
    // Epilogue
#pragma unroll
    for (int j = 0; j < 4; ++j)
        acc[j] = __builtin_amdgcn_wmma_f32_16x16x4_f32(
                     false, av, false, bv[j], (short)0, acc[j], false, false);

#pragma unroll
    for (int j = 0; j < 4; ++j)
#pragma unroll
        for (int v = 0; v < 8; ++v)
            Cb[(long long)(half * 8 + v) * ldc + j * 16 + idx] = acc[j][v];
}

// ---------------------------------------------------------------------------
// energies[b][t][e] = v . tanh(pt[t*B+b][:] + pe[e*B+b][:] + b_attn) + b_score
// masked (seq_mask[e][b]) -> -1e12.  One block per (b,t); pt row + v staged
// in LDS; 8 waves sweep e with wave-shuffle reductions over H.
// ---------------------------------------------------------------------------
__global__ void energies_kernel(const float* __restrict__ pt,
                                const float* __restrict__ pe,
                                const float* __restrict__ b_attn,
                                const float* __restrict__ v_score,
                                const float* __restrict__ b_score,
                                const unsigned char* __restrict__ seq_mask,
                                float* __restrict__ energies)
{
    int bt = blockIdx.x;            // 0 .. B*T-1
    int b  = bt / T_LEN;
    int t  = bt % T_LEN;

    __shared__ float s_pt[HID];
    __shared__ float s_v[HID];
    for (int h = threadIdx.x; h < HID; h += blockDim.x) {
        s_pt[h] = pt[((long long)(t * BATCH + b)) * HID + h] + b_attn[h];
        s_v[h]  = v_score[h];
    }
    __syncthreads();

    int wave = threadIdx.x >> 5;
    int lane = threadIdx.x & 31;
    int nw   = blockDim.x >> 5;
    float bs = b_score[0];

    for (int e = wave; e < E_LEN; e += nw) {
        const float* per = pe + ((long long)(e * BATCH + b)) * HID;
        float acc = 0.f;
#pragma unroll
        for (int i = 0; i < HID / 32; ++i) {
            int h = i * 32 + lane;
            acc += tanhf(s_pt[h] + per[h]) * s_v[h];
        }
        for (int off = 16; off; off >>= 1)
            acc += __shfl_xor(acc, off, 32);
        if (lane == 0) {
            float val = seq_mask[(long long)e * BATCH + b] ? -1e12f : (acc + bs);
            energies[(long long)bt * E_LEN + e] = val;
        }
    }
}

// softmax over E (axis 2 of [B,T,E]); one wave per (b,t) row.
__global__ void softmax_e_kernel(const float* __restrict__ energies,
                                 float* __restrict__ attn_s2)
{
    int row = (int)((blockIdx.x * blockDim.x + threadIdx.x) >> 5);
    if (row >= BATCH * T_LEN) return;
    int lane = threadIdx.x & 31;
    const float* er = energies + (long long)row * E_LEN;

    float vals[E_LEN / 32];
    float m = -3.4e38f;
#pragma unroll
    for (int i = 0; i < E_LEN / 32; ++i) {
        vals[i] = er[i * 32 + lane];
        m = fmaxf(m, vals[i]);
    }
    for (int off = 16; off; off >>= 1)
        m = fmaxf(m, __shfl_xor(m, off, 32));
    float s = 0.f;
#pragma unroll
    for (int i = 0; i < E_LEN / 32; ++i) {
        vals[i] = __expf(vals[i] - m);
        s += vals[i];
    }
    for (int off = 16; off; off >>= 1)
        s += __shfl_xor(s, off, 32);
    float inv = 1.f / s;
#pragma unroll
    for (int i = 0; i < E_LEN / 32; ++i)
        attn_s2[(long long)row * E_LEN + i * 32 + lane] = vals[i] * inv;
}

// softmax over T (axis 2 of [B,E,T] after transpose); one wave per (b,e),
// lane == t (T_LEN == 32 == wave width).
__global__ void softmax_t_kernel(const float* __restrict__ energies,
                                 float* __restrict__ attn_s1)
{
    int w = (int)((blockIdx.x * blockDim.x + threadIdx.x) >> 5);
    if (w >= BATCH * E_LEN) return;
    int b = w / E_LEN;
    int e = w % E_LEN;
    int lane = threadIdx.x & 31;   // t

    float val = energies[((long long)(b * T_LEN + lane)) * E_LEN + e];
    float m = val;
    for (int off = 16; off; off >>= 1)
        m = fmaxf(m, __shfl_xor(m, off, 32));
    float x = __expf(val - m);
    float s = x;
    for (int off = 16; off; off >>= 1)
        s += __shfl_xor(s, off, 32);
    attn_s1[((long long)(b * E_LEN + e)) * T_LEN + lane] = x / s;
}

// ---------------------------------------------------------------------------
extern "C" void kernel_launch(void* const* d_in, const int* in_sizes, int n_in,
                              void* d_out, int out_size, void* d_ws, size_t ws_size,
                              hipStream_t stream)
{
    const float*         topic    = (const float*)d_in[0];          // [T,B,D]
    const float*         exp_out  = (const float*)d_in[1];          // [E,B,D]
    const unsigned char* seq_mask = (const unsigned char*)d_in[2];  // [E,B] bool
    const float*         W_attn   = (const float*)d_in[3];          // [H, H+D]
    const float*         b_attn   = (const float*)d_in[4];          // [H]
    const float*         v_score  = (const float*)d_in[5];          // [1,H]
    const float*         b_score  = (const float*)d_in[6];          // [1]
    float*               out      = (float*)d_out;                  // new_topic | new_exp

    // Workspace layout (floats)
    float* ws       = (float*)d_ws;
    float* pt       = ws;                                  // [T*B, H]   262144
    float* pe       = pt + (long long)T_LEN * BATCH * HID; // [E*B, H]  2097152
    float* energies = pe + (long long)E_LEN * BATCH * HID; // [B,T,E]    131072
    float* s1       = energies + (long long)BATCH * T_LEN * E_LEN; // [B,E,T]
    float* s2       = s1       + (long long)BATCH * E_LEN * T_LEN; // [B,T,E]

    float* new_topic = out;
    float* new_exp   = out + (long long)BATCH * T_LEN * DIM;

    const int THREADS = 256;              // 8 waves per block

    // 1) pt[(t*B+b)][h] = sum_d topic[(t*B+b)*D+d] * W_attn[h*(H+D)+d]
    {
        int tilesM = (T_LEN * BATCH) / 16, tilesN4 = HID / 64;
        int waves = tilesM * tilesN4;
        wmma_gemm_f32<1><<<(waves * 32 + THREADS - 1) / THREADS, THREADS, 0, stream>>>(
            topic, 0, DIM,
            W_attn, 0, HID + DIM, /*koff=*/0,
            pt, 0, HID,
            tilesM, tilesN4, DIM, 1);
    }
    // 2) pe[(e*B+b)][h] = sum_d exp[(e*B+b)*D+d] * W_attn[h*(H+D)+D+d]
    {
        int tilesM = (E_LEN * BATCH) / 16, tilesN4 = HID / 64;
        int waves = tilesM * tilesN4;
        wmma_gemm_f32<1><<<(waves * 32 + THREADS - 1) / THREADS, THREADS, 0, stream>>>(
            exp_out, 0, DIM,
            W_attn, 0, HID + DIM, /*koff=*/DIM,
            pe, 0, HID,
            tilesM, tilesN4, DIM, 1);
    }
    // 3) energies [B,T,E] with tanh + mask
    energies_kernel<<<BATCH * T_LEN, THREADS, 0, stream>>>(
        pt, pe, b_attn, v_score, b_score, seq_mask, energies);

    // 4) softmax over E -> s2 [B,T,E]
    softmax_e_kernel<<<(BATCH * T_LEN * 32 + THREADS - 1) / THREADS, THREADS, 0, stream>>>(
        energies, s2);
    // 5) softmax over T -> s1 [B,E,T]
    softmax_t_kernel<<<(BATCH * E_LEN * 32 + THREADS - 1) / THREADS, THREADS, 0, stream>>>(
        energies, s1);

    // 6) new_topic[b][t][d] = sum_e s2[b][t][e] * exp[e*B*D + b*D + d]
    {
        int tilesM = T_LEN / 16, tilesN4 = DIM / 64;
        int waves = tilesM * tilesN4 * BATCH;
        wmma_gemm_f32<0><<<(waves * 32 + THREADS - 1) / THREADS, THREADS, 0, stream>>>(
            s2, (long long)T_LEN * E_LEN, E_LEN,
            exp_out, (long long)DIM /*b*D*/, BATCH * DIM, /*koff=*/0,
            new_topic, (long long)T_LEN * DIM, DIM,
            tilesM, tilesN4, E_LEN, BATCH);
    }
    // 7) new_exp[b][e][d] = sum_t s1[b][e][t] * topic[t*B*D + b*D + d]
    {
        int tilesM = E_LEN / 16, tilesN4 = DIM / 64;
        int waves = tilesM * tilesN4 * BATCH;
        wmma_gemm_f32<0><<<(waves * 32 + THREADS - 1) / THREADS, THREADS, 0, stream>>>(
            s1, (long long)E_LEN * T_LEN, T_LEN,
            topic, (long long)DIM, BATCH * DIM, /*koff=*/0,
            new_exp, (long long)E_LEN * DIM, DIM,
            tilesM, tilesN4, T_LEN, BATCH);
    }
}